// VanilaGCN_77446850282016
// MI455X (gfx1250) — compile-verified
//
#include <hip/hip_runtime.h>

typedef float v2f __attribute__((ext_vector_type(2)));
typedef float v8f __attribute__((ext_vector_type(8)));

#define N_NODES   50000
#define N_EDGES   800000
#define IN_F      128
#define CONV_H    96
#define LIN_H     64
#define N_CLASSES 10
#define N_GRAPHS  128

// ---------------------------------------------------------------------------
// Elementwise / graph-prep kernels
// ---------------------------------------------------------------------------
__global__ void k_fill(float* __restrict__ p, float v, int n) {
    int i = blockIdx.x * blockDim.x + threadIdx.x;
    if (i < n) p[i] = v;
}

__global__ void k_degree(const int* __restrict__ col, const float* __restrict__ w,
                         float* __restrict__ deg, int E) {
    int e = blockIdx.x * blockDim.x + threadIdx.x;
    if (e < E) atomicAdd(&deg[col[e]], w[e]);
}

__global__ void k_dinv(float* __restrict__ deg, int n) {
    int i = blockIdx.x * blockDim.x + threadIdx.x;
    if (i < n) {
        float d = deg[i];
        deg[i] = (d > 0.0f) ? rsqrtf(d) : 0.0f;
    }
}

// norm[e] = dinv[row]*w*dinv[col] for e<E ; norm[E+n] = dinv[n]^2 (self loops)
__global__ void k_norm(const int* __restrict__ row, const int* __restrict__ col,
                       const float* __restrict__ w, const float* __restrict__ dinv,
                       float* __restrict__ norm, int E, int N) {
    int i = blockIdx.x * blockDim.x + threadIdx.x;
    if (i < E) {
        norm[i] = dinv[row[i]] * w[i] * dinv[col[i]];
    } else if (i < E + N) {
        float d = dinv[i - E];
        norm[i] = d * d;
    }
}

// ---------------------------------------------------------------------------
// FP32 WMMA GEMM:  C[M,NCOLS] = A[M,K] @ B[K,NCOLS]   (compile-time K, NCOLS)
// Block = 256 threads = 8 waves; blockIdx.y picks the 16-wide n-tile, the
// 8 waves cover 8 consecutive 16-row m-tiles. B tile (K x 16) is staged once
// into LDS, pair-interleaved so each lane's {k,k+1} pair is one ds_load_b64
// at an immediate offset. Fully unrolled K loop -> v_wmma_f32_16x16x4_f32
// with immediate-offset global_load_b64 for A.
//   A frag (16x4): lanes 0-15 -> K={k,k+1}, lanes 16-31 -> K={k+2,k+3}, M=lane&15
//   B frag (4x16): same half-wave K split, N=lane&15
//   C/D (16x16):   VGPR j -> M = j + 8*(lane>>4), N = lane&15
// ---------------------------------------------------------------------------
template <int K, int NCOLS>
__global__ void k_gemm_wmma_f32(const float* __restrict__ A, const float* __restrict__ B,
                                float* __restrict__ C, int M) {
    __shared__ float bT[K * 16];            // [(k>>1)*32 + c*2 + (k&1)]
    const int n0 = blockIdx.y << 4;

    // stage B tile: coalesced global read, pair-interleaved LDS write
#pragma unroll
    for (int j = 0; j < (K * 16) / 256; ++j) {
        int idx = threadIdx.x + j * 256;
        int kk = idx >> 4, c = idx & 15;
        bT[(kk >> 1) * 32 + c * 2 + (kk & 1)] = B[kk * NCOLS + n0 + c];
    }
    __syncthreads();

    const int lane = threadIdx.x & 31;
    const int wave = threadIdx.x >> 5;
    const int tm = blockIdx.x * 8 + wave;
    if (tm >= (M >> 4)) return;             // wave-uniform; after barrier

    const int m0 = tm << 4;
    const int half = lane >> 4;             // 0: K pair {0,1}; 1: K pair {2,3}
    const int r = lane & 15;

    const float* aRow = A + (size_t)(m0 + r) * K + 2 * half;
    const v2f* bp = (const v2f*)bT + (half * 16 + r);   // + (k/2)*16 per step

    v8f acc = {};
#pragma unroll
    for (int k = 0; k < K; k += 4) {
        v2f a = *(const v2f*)(aRow + k);    // global_load_b64, imm offset
        v2f b = bp[(k >> 1) * 16];          // ds_load_b64, imm offset
        acc = __builtin_amdgcn_wmma_f32_16x16x4_f32(
            /*neg_a=*/false, a, /*neg_b=*/false, b,
            /*c_mod=*/(short)0, acc, /*reuse_a=*/false, /*reuse_b=*/false);
    }

    float* cRow = C + (size_t)(m0 + 8 * half) * NCOLS + n0 + r;
#pragma unroll
    for (int j = 0; j < 8; ++j) cRow[j * NCOLS] = acc[j];
}

// ---------------------------------------------------------------------------
// Aggregation kernels (F compile-time; F%32==0 -> per-wave uniform node/edge id)
// ---------------------------------------------------------------------------
// out[n,f] = y[n,f]*norm_self[n] + bias[f]   (self-loop term + bias, no atomics)
template <int F>
__global__ void k_self_bias(const float* __restrict__ y, const float* __restrict__ normSelf,
                            const float* __restrict__ bias, float* __restrict__ out, int N) {
    int i = blockIdx.x * blockDim.x + threadIdx.x;
    if (i >= N * F) return;
    int n = __builtin_amdgcn_readfirstlane(i / F);     // wave-uniform -> scalar load
    int f = i % F;
    out[i] = y[i] * normSelf[n] + bias[f];
}

// out[col[e], f] += y[row[e], f] * norm[e]   (96 threads = 3 full waves per edge)
template <int F>
__global__ void k_scatter(const int* __restrict__ row, const int* __restrict__ col,
                          const float* __restrict__ norm, const float* __restrict__ y,
                          float* __restrict__ out, int E) {
    long long i = (long long)blockIdx.x * blockDim.x + threadIdx.x;
    if (i >= (long long)E * F) return;
    int e = __builtin_amdgcn_readfirstlane((int)(i / F));   // uniform: s_load row/col/norm
    int f = (int)(i % F);
    float v = y[(size_t)row[e] * F + f] * norm[e];
    atomicAdd(&out[(size_t)col[e] * F + f], v);
}

__global__ void k_relu(float* __restrict__ p, int n) {
    int i = blockIdx.x * blockDim.x + threadIdx.x;
    if (i < n) p[i] = fmaxf(p[i], 0.0f);
}

// g[batch[n], f] = max(...)  — h >= 0 after relu, so unsigned bit-pattern
// atomicMax is order-equivalent to float max with g initialized to 0.0f.
template <int F>
__global__ void k_pool_max(const float* __restrict__ h, const int* __restrict__ batch,
                           unsigned* __restrict__ g, int N) {
    int i = blockIdx.x * blockDim.x + threadIdx.x;
    if (i >= N * F) return;
    int n = __builtin_amdgcn_readfirstlane(i / F);
    int f = i % F;
    atomicMax(&g[(size_t)batch[n] * F + f], __float_as_uint(h[i]));
}

template <int F>
__global__ void k_bias_relu(float* __restrict__ p, const float* __restrict__ bias, int n) {
    int i = blockIdx.x * blockDim.x + threadIdx.x;
    if (i < n) p[i] = fmaxf(p[i] + bias[i % F], 0.0f);
}

// out[g,c] = t[g,:] @ Wl2[:,c] + bl2[c]     (tiny: 128x10, K=64 — scalar)
__global__ void k_lin2(const float* __restrict__ t, const float* __restrict__ W,
                       const float* __restrict__ b, float* __restrict__ out) {
    int i = blockIdx.x * blockDim.x + threadIdx.x;
    if (i >= N_GRAPHS * N_CLASSES) return;
    int g = i / N_CLASSES, c = i % N_CLASSES;
    float s = b[c];
#pragma unroll 8
    for (int k = 0; k < LIN_H; ++k) s += t[g * LIN_H + k] * W[k * N_CLASSES + c];
    out[i] = s;
}

// ---------------------------------------------------------------------------
// Host launch
// ---------------------------------------------------------------------------
static inline int cdiv(long long a, int b) { return (int)((a + b - 1) / b); }

extern "C" void kernel_launch(void* const* d_in, const int* in_sizes, int n_in,
                              void* d_out, int out_size, void* d_ws, size_t ws_size,
                              hipStream_t stream) {
    const float* x   = (const float*)d_in[0];
    const int*   ei  = (const int*)  d_in[1];   // [2, E] flat: row then col
    const float* ew  = (const float*)d_in[2];
    const int*   bat = (const int*)  d_in[3];
    const float* W1  = (const float*)d_in[4];   const float* b1  = (const float*)d_in[5];
    const float* W2  = (const float*)d_in[6];   const float* b2  = (const float*)d_in[7];
    const float* W3  = (const float*)d_in[8];   const float* b3  = (const float*)d_in[9];
    const float* Wl1 = (const float*)d_in[10];  const float* bl1 = (const float*)d_in[11];
    const float* Wl2 = (const float*)d_in[12];  const float* bl2 = (const float*)d_in[13];

    const int N = N_NODES, E = N_EDGES;
    const int* row = ei;
    const int* col = ei + E;

    // workspace layout (floats)
    float* deg  = (float*)d_ws;                 // N
    float* norm = deg  + N;                     // E + N
    float* y    = norm + (E + N);               // N * CONV_H
    float* h    = y    + (size_t)N * CONV_H;    // N * CONV_H
    float* g    = h    + (size_t)N * CONV_H;    // N_GRAPHS * CONV_H
    float* t    = g    + N_GRAPHS * CONV_H;     // N_GRAPHS * LIN_H

    const int T = 256;
    const long long NF = (long long)N * CONV_H;
    const long long EF = (long long)E * CONV_H;

    // --- normalization coefficients (computed once, reused for all 3 layers)
    k_fill  <<<cdiv(N, T), T, 0, stream>>>(deg, 1.0f, N);        // self-loop weight
    k_degree<<<cdiv(E, T), T, 0, stream>>>(col, ew, deg, E);
    k_dinv  <<<cdiv(N, T), T, 0, stream>>>(deg, N);
    k_norm  <<<cdiv(E + N, T), T, 0, stream>>>(row, col, ew, deg, norm, E, N);

    // --- layer 1: GEMM (WMMA f32, K=128) -> self+bias -> edge scatter -> relu
    {
        dim3 grid(cdiv(N / 16, 8), CONV_H / 16);
        k_gemm_wmma_f32<IN_F, CONV_H><<<grid, T, 0, stream>>>(x, W1, y, N);
        k_self_bias<CONV_H><<<cdiv(NF, T), T, 0, stream>>>(y, norm + E, b1, h, N);
        k_scatter<CONV_H><<<cdiv(EF, T), T, 0, stream>>>(row, col, norm, y, h, E);
        k_relu<<<cdiv(NF, T), T, 0, stream>>>(h, (int)NF);
    }
    // --- layers 2 & 3: same with K=96
    const float* Ws[2] = {W2, W3};
    const float* bs[2] = {b2, b3};
    for (int l = 0; l < 2; ++l) {
        dim3 grid(cdiv(N / 16, 8), CONV_H / 16);
        k_gemm_wmma_f32<CONV_H, CONV_H><<<grid, T, 0, stream>>>(h, Ws[l], y, N);
        k_self_bias<CONV_H><<<cdiv(NF, T), T, 0, stream>>>(y, norm + E, bs[l], h, N);
        k_scatter<CONV_H><<<cdiv(EF, T), T, 0, stream>>>(row, col, norm, y, h, E);
        k_relu<<<cdiv(NF, T), T, 0, stream>>>(h, (int)NF);
    }

    // --- global max pool over graphs
    k_fill<<<cdiv(N_GRAPHS * CONV_H, T), T, 0, stream>>>(g, 0.0f, N_GRAPHS * CONV_H);
    k_pool_max<CONV_H><<<cdiv(NF, T), T, 0, stream>>>(h, bat, (unsigned*)g, N);

    // --- MLP head: [128,96]@[96,64] via WMMA, relu, then scalar [128,64]@[64,10]
    {
        dim3 grid(1, LIN_H / 16);               // 8 m-tiles = 8 waves of one block
        k_gemm_wmma_f32<CONV_H, LIN_H><<<grid, T, 0, stream>>>(g, Wl1, t, N_GRAPHS);
        k_bias_relu<LIN_H><<<cdiv(N_GRAPHS * LIN_H, T), T, 0, stream>>>(t, bl1, N_GRAPHS * LIN_H);
        k_lin2<<<cdiv(N_GRAPHS * N_CLASSES, T), T, 0, stream>>>(t, Wl2, bl2, (float*)d_out);
    }
}